// FeatureMemory_10917806866871
// MI455X (gfx1250) — compile-verified
//
#include <hip/hip_runtime.h>

// ---------------------------------------------------------------------------
// FeatureMemory on MI455X (gfx1250): linear-attention algebraic collapse.
//   S_f = X_f X_f^T (64x64 Gram per frame, WMMA f32 16x16x4)
//   tiny 64x64 chain folds all 6 linears into Bmat(64x64), d(64)
//   out = memory(64x14400 as 14400x64) @ Bmat + d   (WMMA f32)
// Data movement: async global->LDS (GLOBAL_LOAD_ASYNC_TO_LDS_B128) with
// double buffering where the toolchain exposes it; sync fallback otherwise.
// ---------------------------------------------------------------------------

typedef float v2f __attribute__((ext_vector_type(2)));
typedef float v8f __attribute__((ext_vector_type(8)));

#define HW 14400          // 120*120
#define NB 75             // reduce blocks per frame; 75 * 192 = 14400
#define PIXB 192          // pixels per reduce block (3 tiles of 64)
#define PSTRIDE (64*64 + 64)

#if __has_builtin(__builtin_amdgcn_global_load_async_to_lds_b128)
#define HAVE_ASYNC_LDS 1
#else
#define HAVE_ASYNC_LDS 0
#endif

// async builtin wants v4i pointers: global (AS1) src, LDS (AS3) dst
typedef int v4i __attribute__((vector_size(16)));
typedef __attribute__((address_space(1))) v4i gv4i;
typedef __attribute__((address_space(3))) v4i lv4i;

// 16B global -> LDS copy: async path if available, else through VGPRs.
__device__ __forceinline__ void copy16_g2l(float* lds_dst, const float* g_src) {
#if HAVE_ASYNC_LDS
  __builtin_amdgcn_global_load_async_to_lds_b128((gv4i*)g_src, (lv4i*)lds_dst,
                                                 /*offset=*/0, /*cpol=*/0);
#else
  *(float4*)lds_dst = *(const float4*)g_src;
#endif
}

template <int N>
__device__ __forceinline__ void async_wait() {
#if HAVE_ASYNC_LDS
#if __has_builtin(__builtin_amdgcn_s_wait_asynccnt)
  __builtin_amdgcn_s_wait_asynccnt(N);
#else
  asm volatile("s_wait_asynccnt %0" ::"i"(N) : "memory");
#endif
#endif
}

__device__ __forceinline__ v8f wmma4(v2f a, v2f b, v8f c) {
  // D = A(16x4 f32) x B(4x16 f32) + C(16x16 f32)
  return __builtin_amdgcn_wmma_f32_16x16x4_f32(
      /*neg_a=*/false, a, /*neg_b=*/false, b,
      /*c_mod=*/(short)0, c, /*reuse_a=*/false, /*reuse_b=*/false);
}

__device__ __forceinline__ v8f vzero8() {
  v8f z = {0.f, 0.f, 0.f, 0.f, 0.f, 0.f, 0.f, 0.f};
  return z;
}

// ---------------------------------------------------------------------------
// Kernel 1: per-frame partial Gram S = X X^T and channel sums s = X 1.
// grid = (NB, 2 frames), block = 256 (8 waves). Each wave owns 2 of the 16
// 16x16 output tiles. Tiles stream through a double-buffered LDS stage fed
// by async global->LDS loads.
// ---------------------------------------------------------------------------
__global__ void k_gram(const float* __restrict__ feats, float* __restrict__ part) {
  const int f    = blockIdx.y;
  const int b    = blockIdx.x;
  const int t    = threadIdx.x;
  const int wave = t >> 5;
  const int lane = t & 31;
  const int half = lane >> 4;
  const int l15  = lane & 15;
  const int tile0 = wave * 2;

  __shared__ float X[2][64][64];   // double-buffered [channel][pixel-local]

  v8f acc0 = vzero8();
  v8f acc1 = vzero8();
  float ssum = 0.f;

  const float* base = feats + (size_t)f * 64 * HW;
  const int p_start = b * PIXB;

  auto issue_tile = [&](int buf, int p0) {
    #pragma unroll
    for (int it = 0; it < 4; ++it) {
      const int idx  = it * 256 + t;       // 0..1023 float4s
      const int row  = idx >> 4;
      const int col4 = idx & 15;
      copy16_g2l(&X[buf][row][col4 * 4],
                 base + (size_t)row * HW + p0 + col4 * 4);
    }
  };

  issue_tile(0, p_start);
  for (int tp = 0; tp < 3; ++tp) {
    if (tp < 2) issue_tile((tp + 1) & 1, p_start + (tp + 1) * 64);
    // in-order async completion: <=4 outstanding means current tile landed
    if (tp < 2) async_wait<4>(); else async_wait<0>();
    __syncthreads();

    const float (*Xt)[64] = X[tp & 1];

    if (t < 64) {   // channel sums (s = X 1)
      float s = 0.f;
      #pragma unroll
      for (int p = 0; p < 64; ++p) s += Xt[t][p];
      ssum += s;
    }

    // S += Xtile * Xtile^T via v_wmma_f32_16x16x4_f32
    #pragma unroll
    for (int k0 = 0; k0 < 64; k0 += 4) {
      const int k = k0 + 2 * half;
      const int mi0 = (tile0)     >> 2, ni0 = (tile0)     & 3;
      const int mi1 = (tile0 + 1) >> 2, ni1 = (tile0 + 1) & 3;
      v2f a0, b0, a1, b1;
      a0.x = Xt[mi0 * 16 + l15][k];  a0.y = Xt[mi0 * 16 + l15][k + 1];
      b0.x = Xt[ni0 * 16 + l15][k];  b0.y = Xt[ni0 * 16 + l15][k + 1];
      a1.x = Xt[mi1 * 16 + l15][k];  a1.y = Xt[mi1 * 16 + l15][k + 1];
      b1.x = Xt[ni1 * 16 + l15][k];  b1.y = Xt[ni1 * 16 + l15][k + 1];
      acc0 = wmma4(a0, b0, acc0);
      acc1 = wmma4(a1, b1, acc1);
    }
    __syncthreads();   // all reads done before next issue overwrites buffer
  }

  float* dst = part + (size_t)(f * NB + b) * PSTRIDE;
  {
    const int mi = tile0 >> 2, ni = tile0 & 3;
    #pragma unroll
    for (int r = 0; r < 8; ++r)
      dst[(mi * 16 + r + 8 * half) * 64 + ni * 16 + l15] = acc0[r];
  }
  {
    const int mi = (tile0 + 1) >> 2, ni = (tile0 + 1) & 3;
    #pragma unroll
    for (int r = 0; r < 8; ++r)
      dst[(mi * 16 + r + 8 * half) * 64 + ni * 16 + l15] = acc1[r];
  }
  if (t < 64) dst[4096 + t] = ssum;
}

// ---------------------------------------------------------------------------
// LDS 64x64 matmul (D = op(A) * op(B)) with WMMA; block = 256 threads.
// ---------------------------------------------------------------------------
__device__ void mm64(float (*D)[64], const float (*A)[64], const float (*B)[64],
                     bool tA, bool tB, int t) {
  const int wave = t >> 5, lane = t & 31, half = lane >> 4, l15 = lane & 15;
  const int tile0 = wave * 2;
  v8f acc[2];
  acc[0] = vzero8();
  acc[1] = vzero8();
  __syncthreads();
  for (int k0 = 0; k0 < 64; k0 += 4) {
    const int k = k0 + 2 * half;
    #pragma unroll
    for (int q = 0; q < 2; ++q) {
      const int mi = (tile0 + q) >> 2, ni = (tile0 + q) & 3;
      v2f a, bb;
      a.x  = tA ? A[k][mi * 16 + l15]     : A[mi * 16 + l15][k];
      a.y  = tA ? A[k + 1][mi * 16 + l15] : A[mi * 16 + l15][k + 1];
      bb.x = tB ? B[ni * 16 + l15][k]     : B[k][ni * 16 + l15];
      bb.y = tB ? B[ni * 16 + l15][k + 1] : B[k + 1][ni * 16 + l15];
      acc[q] = wmma4(a, bb, acc[q]);
    }
  }
  __syncthreads();
  #pragma unroll
  for (int q = 0; q < 2; ++q) {
    const int mi = (tile0 + q) >> 2, ni = (tile0 + q) & 3;
    #pragma unroll
    for (int r = 0; r < 8; ++r)
      D[mi * 16 + r + 8 * half][ni * 16 + l15] = acc[q][r];
  }
  __syncthreads();
}

__device__ void ldw(float (*Dst)[64], const float* __restrict__ g, int t) {
  #pragma unroll
  for (int i = 0; i < 16; ++i) {
    const int e = i * 256 + t;
    Dst[e >> 6][e & 63] = g[e];
  }
  __syncthreads();
}

// ---------------------------------------------------------------------------
// Kernel 2: reduce partials, then fold everything into Bmat / d.
//   KtV  = wi2 S wi3^T + (wi2 s) bi3^T + bi2 (wi3 s)^T + N bi2 bi3^T
//   Amat = wi1^T KtV ; c = KtV^T bi1
//   G    = Amat^T S0 Amat + r c^T + c r^T + HW c c^T ; r = Amat^T s0
//   g    = r + HW c
//   KV2  = w1 G w2^T + (w1 g) b2^T + b1 (w2 g)^T + HW b1 b2^T
//   Bmat = w3^T KV2 ; d = KV2^T b3
// ---------------------------------------------------------------------------
__global__ void k_combine(const float* __restrict__ part,
    const float* __restrict__ w1,  const float* __restrict__ b1,
    const float* __restrict__ w2,  const float* __restrict__ b2,
    const float* __restrict__ w3,  const float* __restrict__ b3,
    const float* __restrict__ wi1, const float* __restrict__ bi1,
    const float* __restrict__ wi2, const float* __restrict__ bi2,
    const float* __restrict__ wi3, const float* __restrict__ bi3,
    float* __restrict__ wsS0, float* __restrict__ wsB, float* __restrict__ wsd) {
  __shared__ float P[64][64], Q[64][64], R[64][64];
  __shared__ float s_all[64], s_0[64], u[64], w[64], cv[64], rv[64], gv[64];
  const int t = threadIdx.x;

  // deterministic partial reduction: S_all -> P (LDS), S_0 -> global ws
  #pragma unroll 2
  for (int i = 0; i < 16; ++i) {
    const int e = i * 256 + t;
    float v0 = 0.f, v1 = 0.f;
    for (int b = 0; b < NB; ++b) {
      v0 += part[(size_t)(0 * NB + b) * PSTRIDE + e];
      v1 += part[(size_t)(1 * NB + b) * PSTRIDE + e];
    }
    P[e >> 6][e & 63] = v0 + v1;
    wsS0[e] = v0;
  }
  if (t < 64) {
    float v0 = 0.f, v1 = 0.f;
    for (int b = 0; b < NB; ++b) {
      v0 += part[(size_t)(0 * NB + b) * PSTRIDE + 4096 + t];
      v1 += part[(size_t)(1 * NB + b) * PSTRIDE + 4096 + t];
    }
    s_0[t] = v0; s_all[t] = v0 + v1;
  }
  __syncthreads();

  ldw(Q, wi2, t);                                           // Q = wi2
  if (t < 64) { float s = 0.f; for (int k = 0; k < 64; ++k) s += Q[t][k] * s_all[k]; u[t] = s; }
  __syncthreads();
  mm64(R, Q, P, false, false, t);                           // R = wi2 S
  ldw(P, wi3, t);                                           // P = wi3
  if (t < 64) { float s = 0.f; for (int k = 0; k < 64; ++k) s += P[t][k] * s_all[k]; w[t] = s; }
  __syncthreads();
  mm64(Q, R, P, false, true, t);                            // Q = wi2 S wi3^T
  #pragma unroll 2
  for (int i = 0; i < 16; ++i) {                            // + rank-1 terms (N=28800)
    const int e = i * 256 + t, r_ = e >> 6, c_ = e & 63;
    Q[r_][c_] += u[r_] * bi3[c_] + bi2[r_] * w[c_] + 28800.f * bi2[r_] * bi3[c_];
  }
  __syncthreads();
  ldw(P, wi1, t);                                           // P = wi1
  mm64(R, P, Q, true, false, t);                            // R = Amat = wi1^T KtV
  if (t < 64) { float s = 0.f; for (int k = 0; k < 64; ++k) s += bi1[k] * Q[k][t]; cv[t] = s; }
  __syncthreads();
  #pragma unroll 2
  for (int i = 0; i < 16; ++i) { const int e = i * 256 + t; P[e >> 6][e & 63] = wsS0[e]; }
  __syncthreads();                                          // P = S_0
  mm64(Q, R, P, true, false, t);                            // Q = Amat^T S_0
  if (t < 64) { float s = 0.f; for (int k = 0; k < 64; ++k) s += R[k][t] * s_0[k]; rv[t] = s; }
  __syncthreads();
  mm64(P, Q, R, false, false, t);                           // P = G
  #pragma unroll 2
  for (int i = 0; i < 16; ++i) {
    const int e = i * 256 + t, r_ = e >> 6, c_ = e & 63;
    P[r_][c_] += rv[r_] * cv[c_] + cv[r_] * rv[c_] + 14400.f * cv[r_] * cv[c_];
  }
  if (t < 64) gv[t] = rv[t] + 14400.f * cv[t];
  __syncthreads();
  ldw(Q, w1, t);                                            // Q = w1
  if (t < 64) { float s = 0.f; for (int k = 0; k < 64; ++k) s += Q[t][k] * gv[k]; u[t] = s; }
  __syncthreads();
  mm64(R, Q, P, false, false, t);                           // R = w1 G
  ldw(P, w2, t);                                            // P = w2
  if (t < 64) { float s = 0.f; for (int k = 0; k < 64; ++k) s += P[t][k] * gv[k]; w[t] = s; }
  __syncthreads();
  mm64(Q, R, P, false, true, t);                            // Q = w1 G w2^T
  #pragma unroll 2
  for (int i = 0; i < 16; ++i) {
    const int e = i * 256 + t, r_ = e >> 6, c_ = e & 63;
    Q[r_][c_] += u[r_] * b2[c_] + b1[r_] * w[c_] + 14400.f * b1[r_] * b2[c_];
  }
  __syncthreads();
  ldw(P, w3, t);                                            // P = w3
  mm64(R, P, Q, true, false, t);                            // R = Bmat = w3^T KV2
  if (t < 64) { float s = 0.f; for (int k = 0; k < 64; ++k) s += b3[k] * Q[k][t]; gv[t] = s; }
  __syncthreads();
  #pragma unroll 2
  for (int i = 0; i < 16; ++i) { const int e = i * 256 + t; wsB[e] = R[e >> 6][e & 63]; }
  if (t < 64) wsd[t] = gv[t];
}

// ---------------------------------------------------------------------------
// Kernel 3: out[c][p] = sum_k memory[k][p] * Bmat[k][c] + d[c]
// block = 256 (8 waves x 16-pixel tiles = 128 pixels); WMMA over K=64.
// Results staged through LDS for coalesced channel-major stores.
// ---------------------------------------------------------------------------
__global__ void k_apply(const float* __restrict__ mem, const float* __restrict__ Bm,
                        const float* __restrict__ dv, float* __restrict__ out) {
  __shared__ float Bs[64][64];
  __shared__ float Os[64][128];
  __shared__ float ds[64];
  const int t = threadIdx.x, wave = t >> 5, lane = t & 31;
  const int half = lane >> 4, l15 = lane & 15;

  #pragma unroll
  for (int i = 0; i < 4; ++i) {          // Bmat -> LDS via async path
    const int idx = i * 256 + t;         // 0..1023 float4s
    copy16_g2l(&Bs[idx >> 4][(idx & 15) * 4], Bm + idx * 4);
  }
  if (t < 64) ds[t] = dv[t];
  async_wait<0>();
  __syncthreads();

  const int p0 = blockIdx.x * 128 + wave * 16;
  if (p0 < HW) {                      // wave-uniform: EXEC stays all-ones for WMMA
    v8f acc[4];
    #pragma unroll
    for (int q = 0; q < 4; ++q) acc[q] = vzero8();
    #pragma unroll 4
    for (int k0 = 0; k0 < 64; k0 += 4) {
      const int k = k0 + 2 * half;
      v2f a;
      a.x = mem[(size_t)(k)     * HW + p0 + l15];
      a.y = mem[(size_t)(k + 1) * HW + p0 + l15];
      #pragma unroll
      for (int q = 0; q < 4; ++q) {
        v2f b;
        b.x = Bs[k][q * 16 + l15];
        b.y = Bs[k + 1][q * 16 + l15];
        acc[q] = wmma4(a, b, acc[q]);
      }
    }
    #pragma unroll
    for (int q = 0; q < 4; ++q)
      #pragma unroll
      for (int r = 0; r < 8; ++r)
        Os[q * 16 + l15][wave * 16 + r + 8 * half] = acc[q][r];
  }
  __syncthreads();

  const int pblock = blockIdx.x * 128;
  const int nvalid = (HW - pblock < 128) ? (HW - pblock) : 128;
  #pragma unroll 2
  for (int i = 0; i < 8; ++i) {       // float4 coalesced stores
    const int e = i * 256 + t;        // 0..2047 float4s
    const int ch = e >> 5;
    const int px = (e & 31) * 4;
    if (px < nvalid) {
      float4 v;
      v.x = Os[ch][px]     + ds[ch];
      v.y = Os[ch][px + 1] + ds[ch];
      v.z = Os[ch][px + 2] + ds[ch];
      v.w = Os[ch][px + 3] + ds[ch];
      *(float4*)(out + (size_t)ch * HW + pblock + px) = v;
    }
  }
}

// ---------------------------------------------------------------------------
extern "C" void kernel_launch(void* const* d_in, const int* in_sizes, int n_in,
                              void* d_out, int out_size, void* d_ws, size_t ws_size,
                              hipStream_t stream) {
  const float* feats  = (const float*)d_in[0];
  const float* memory = (const float*)d_in[1];
  const float* w1  = (const float*)d_in[2];
  const float* b1  = (const float*)d_in[3];
  const float* w2  = (const float*)d_in[4];
  const float* b2  = (const float*)d_in[5];
  const float* w3  = (const float*)d_in[6];
  const float* b3  = (const float*)d_in[7];
  const float* wi1 = (const float*)d_in[8];
  const float* bi1 = (const float*)d_in[9];
  const float* wi2 = (const float*)d_in[10];
  const float* bi2 = (const float*)d_in[11];
  const float* wi3 = (const float*)d_in[12];
  const float* bi3 = (const float*)d_in[13];
  float* out = (float*)d_out;

  float* part = (float*)d_ws;                          // 2*NB*PSTRIDE floats
  float* wsS0 = part + (size_t)2 * NB * PSTRIDE;       // 4096
  float* wsB  = wsS0 + 4096;                           // 4096
  float* wsd  = wsB + 4096;                            // 64

  k_gram<<<dim3(NB, 2), 256, 0, stream>>>(feats, part);
  k_combine<<<1, 256, 0, stream>>>(part, w1, b1, w2, b2, w3, b3,
                                   wi1, bi1, wi2, bi2, wi3, bi3,
                                   wsS0, wsB, wsd);
  k_apply<<<(HW + 127) / 128, 256, 0, stream>>>(memory, wsB, wsd, out);
}